// VQSLDSCell_37271726195427
// MI455X (gfx1250) — compile-verified
//
#include <hip/hip_runtime.h>
#include <math.h>

typedef __attribute__((ext_vector_type(2))) float v2f;
typedef __attribute__((ext_vector_type(8))) float v8f;

#define N_TOK 4096
#define DXX   128
#define DZ    64
#define KK    256
#define HH    256
#define KIN   (DZ + DXX)   // 192
#define BETA_F 0.25f

#if __has_builtin(__builtin_amdgcn_global_load_async_to_lds_b32)
#define HAVE_ASYNC_LDS 1
#else
#define HAVE_ASYNC_LDS 0
#endif

__device__ __forceinline__ bool finitef(float v) { return __builtin_isfinite(v); }

// ---------------------------------------------------------------------------
// f32 WMMA: D(16x16,f32) = A(16x4,f32) * B(4x16,f32) + C   (wave32)
// ---------------------------------------------------------------------------
__device__ __forceinline__ v8f wmma_f32_16x16x4(v2f a, v2f b, v8f c) {
  return __builtin_amdgcn_wmma_f32_16x16x4_f32(
      /*neg_a=*/false, a, /*neg_b=*/false, b,
      /*c_mod=*/(short)0, c, /*reuse_a=*/false, /*reuse_b=*/false);
}

// Build X1 = [where(isfinite(z), z, 0) | xt]  -> [N, 192]
__global__ void prep_x1(const float* __restrict__ z, const float* __restrict__ x,
                        float* __restrict__ X1) {
  int i = blockIdx.x * blockDim.x + threadIdx.x;
  if (i >= N_TOK * KIN) return;
  int n = i / KIN, k = i % KIN;
  float v;
  if (k < DZ) { v = z[n * DZ + k]; if (!finitef(v)) v = 0.0f; }
  else        { v = x[n * DXX + (k - DZ)]; }
  X1[i] = v;
}

// Out[M,NTOT] = act(A[M,KTOT] @ W[KTOT,NTOT] + bias).
// One wave computes a 16(M) x 64(N) strip: 4 accumulators share one A frag
// per k-step (4x less A traffic, 4 WMMAs per 9 loads).
template<int KTOT, int NTOT, bool TANH>
__global__ __launch_bounds__(128) void gemm_wmma(
    const float* __restrict__ A, const float* __restrict__ W,
    const float* __restrict__ bias, float* __restrict__ Out) {
  const int wave = blockIdx.x * (blockDim.x >> 5) + (threadIdx.x >> 5);
  const int lane = threadIdx.x & 31;
  const int half = lane >> 4;
  const int l16  = lane & 15;
  const int groupsN = NTOT / 64;            // 4-tile groups along N
  const int tm = wave / groupsN;
  const int tg = wave % groupsN;

  const float* Arow = A + (size_t)(tm * 16 + l16) * KTOT;
  const int col0 = tg * 64 + l16;
  const float* W0 = W + col0;

  v8f c0 = {}, c1 = {}, c2 = {}, c3 = {};
  #pragma unroll 2
  for (int k = 0; k < KTOT; k += 4) {
    const int ks = k + half * 2;
    v2f a; a.x = Arow[ks]; a.y = Arow[ks + 1];
    const float* Wr0 = W0 + (size_t)ks * NTOT;
    const float* Wr1 = W0 + (size_t)(ks + 1) * NTOT;
    v2f b0; b0.x = Wr0[0];  b0.y = Wr1[0];
    v2f b1; b1.x = Wr0[16]; b1.y = Wr1[16];
    v2f b2; b2.x = Wr0[32]; b2.y = Wr1[32];
    v2f b3; b3.x = Wr0[48]; b3.y = Wr1[48];
    c0 = wmma_f32_16x16x4(a, b0, c0);
    c1 = wmma_f32_16x16x4(a, b1, c1);
    c2 = wmma_f32_16x16x4(a, b2, c2);
    c3 = wmma_f32_16x16x4(a, b3, c3);
  }

#define STORE_TILE(CT, TOFF)                                                  \
  {                                                                           \
    const int col = col0 + (TOFF);                                            \
    const float bb = bias[col];                                               \
    _Pragma("unroll")                                                         \
    for (int r = 0; r < 8; ++r) {                                             \
      const int row = tm * 16 + half * 8 + r;                                 \
      float v = (CT)[r] + bb;                                                 \
      if (TANH) v = tanhf(v);                                                 \
      Out[(size_t)row * NTOT + col] = v;                                      \
    }                                                                         \
  }
  STORE_TILE(c0, 0)
  STORE_TILE(c1, 16)
  STORE_TILE(c2, 32)
  STORE_TILE(c3, 48)
#undef STORE_TILE
}

// ---------------------------------------------------------------------------
// threefry2x32 (matches JAX's PRNG)
// ---------------------------------------------------------------------------
__device__ __forceinline__ unsigned rotl32(unsigned x, int r) {
  return (x << r) | (x >> (32 - r));
}
__device__ __forceinline__ void threefry2x32(unsigned k0, unsigned k1,
                                             unsigned c0, unsigned c1,
                                             unsigned& o0, unsigned& o1) {
  const unsigned ks0 = k0, ks1 = k1, ks2 = 0x1BD11BDAu ^ k0 ^ k1;
  unsigned x0 = c0 + ks0, x1 = c1 + ks1;
#define TF_RND(r) { x0 += x1; x1 = rotl32(x1, (r)); x1 ^= x0; }
  TF_RND(13) TF_RND(15) TF_RND(26) TF_RND(6)   x0 += ks1; x1 += ks2 + 1u;
  TF_RND(17) TF_RND(29) TF_RND(16) TF_RND(24)  x0 += ks2; x1 += ks0 + 2u;
  TF_RND(13) TF_RND(15) TF_RND(26) TF_RND(6)   x0 += ks0; x1 += ks1 + 3u;
  TF_RND(17) TF_RND(29) TF_RND(16) TF_RND(24)  x0 += ks1; x1 += ks2 + 4u;
  TF_RND(13) TF_RND(15) TF_RND(26) TF_RND(6)   x0 += ks2; x1 += ks0 + 5u;
#undef TF_RND
  o0 = x0; o1 = x1;
}

__device__ __forceinline__ float gumbel_from_bits(unsigned bits) {
  const float tiny = 1.17549435e-38f;
  union { unsigned u; float f; } cv;
  cv.u = (bits >> 9) | 0x3F800000u;          // [1,2)
  float u01 = cv.f - 1.0f;                   // [0,1)
  float u = u01 * (1.0f - tiny) + tiny;      // [tiny,1)
  u = fmaxf(u, tiny);
  return -logf(-logf(u));
}

// ---------------------------------------------------------------------------
// Fused tail: VQ argmin, one-hot-aware einsum, normalize+select, gumbel-max
// categorical, mask select, z_new, kl / discrete_kl.  One block per sample.
// ---------------------------------------------------------------------------
__global__ __launch_bounds__(KK) void fused_tail(
    const float* __restrict__ gt,        // [N,64]  (ws)
    const float* __restrict__ ksample,   // [N,K]
    const float* __restrict__ T,         // [N,K,K]
    const float* __restrict__ startpk,   // [N,K]
    const unsigned char* __restrict__ mask, // [N] bool
    const float* __restrict__ C,         // [K,64]
    const int* __restrict__ seedp,
    float* __restrict__ out_z,           // [N,64]
    float* __restrict__ out_kl,          // [N]
    float* __restrict__ out_dkl,         // [N]
    float* __restrict__ out_qk) {        // [N,K]
  __shared__ float s_gt[DZ];
  __shared__ float s_kf[KK];
  __shared__ float s_pk[KK];
  __shared__ float s_val[KK];
  __shared__ int   s_idx[KK];
  __shared__ float s_sum;
  __shared__ int   s_qk, s_pkidx;

  const int n = blockIdx.x;
  const int j = threadIdx.x;

#if HAVE_ASYNC_LDS
  // Async global->LDS staging of the per-sample rows (ASYNCcnt path).
  {
    typedef __attribute__((address_space(1))) int gint;
    typedef __attribute__((address_space(3))) int lint;
    __builtin_amdgcn_global_load_async_to_lds_b32(
        (gint*)(ksample + (size_t)n * KK + j), (lint*)&s_kf[j], 0, 0);
    if (j < DZ) {
      __builtin_amdgcn_global_load_async_to_lds_b32(
          (gint*)(gt + (size_t)n * DZ + j), (lint*)&s_gt[j], 0, 0);
    }
    asm volatile("s_wait_asynccnt 0x0" ::: "memory");
  }
  __syncthreads();
  const float ksv = s_kf[j];
  __syncthreads();
  s_kf[j] = finitef(ksv) ? ksv : 1.0f;
#else
  if (j < DZ) s_gt[j] = gt[(size_t)n * DZ + j];
  const float ksv = ksample[(size_t)n * KK + j];
  s_kf[j] = finitef(ksv) ? ksv : 1.0f;
#endif
  __syncthreads();

  // ---- VQ: distance to code j, block argmin (first index on ties) ----
  float d2 = 0.0f;
  const float* Cj = C + (size_t)j * DZ;
  #pragma unroll 8
  for (int d = 0; d < DZ; ++d) { float df = s_gt[d] - Cj[d]; d2 += df * df; }
  s_val[j] = d2; s_idx[j] = j;
  __syncthreads();
  for (int off = KK / 2; off > 0; off >>= 1) {
    if (j < off) {
      float v1 = s_val[j], v2 = s_val[j + off];
      int   i1 = s_idx[j], i2 = s_idx[j + off];
      if (v2 < v1 || (v2 == v1 && i2 < i1)) { s_val[j] = v2; s_idx[j] = i2; }
    }
    __syncthreads();
  }
  if (j == 0) s_qk = s_idx[0];
  __syncthreads();
  const int qk = s_qk;
  out_qk[(size_t)n * KK + j] = (j == qk) ? 1.0f : 0.0f;

  // ---- einsum nk,nkj->nj : uniform branch skips k where kf[k]==0 ----
  const float* Tn = T + (size_t)n * KK * KK + j;
  float acc = 0.0f;
  for (int k = 0; k < KK; ++k) {
    const float w = s_kf[k];                      // uniform across block
    if (w != 0.0f) acc += w * __builtin_nontemporal_load(Tn + (size_t)k * KK);
  }
  s_val[j] = acc;
  __syncthreads();
  for (int off = KK / 2; off > 0; off >>= 1) {
    if (j < off) s_val[j] += s_val[j + off];
    __syncthreads();
  }
  if (j == 0) s_sum = s_val[0];
  __syncthreads();
  float pk = acc / s_sum;
  if (!finitef(ksv)) pk = startpk[(size_t)n * KK + j];
  s_pk[j] = pk;

  // ---- categorical via gumbel-max, threefry keyed as JAX split(key,3)[0] ----
  const unsigned k1 = (unsigned)seedp[0];
  unsigned a0, a1, b0, b1;
  threefry2x32(0u, k1, 0u, 3u, a0, a1);
  threefry2x32(0u, k1, 1u, 4u, b0, b1);
  const unsigned rk0 = a0, rk1 = b0;            // k_rng
  const unsigned NKH = (unsigned)(N_TOK * KK / 2);
  const unsigned i = (unsigned)(n * KK + j);
  unsigned y0, y1, bits;
  if (i < NKH) { threefry2x32(rk0, rk1, i, i + NKH, y0, y1); bits = y0; }
  else         { threefry2x32(rk0, rk1, i - NKH, i, y0, y1); bits = y1; }
  const float logit = logf(pk) + gumbel_from_bits(bits);
  s_val[j] = logit; s_idx[j] = j;
  __syncthreads();
  for (int off = KK / 2; off > 0; off >>= 1) {
    if (j < off) {
      float v1 = s_val[j], v2 = s_val[j + off];
      int   i1 = s_idx[j], i2 = s_idx[j + off];
      if (v2 > v1 || (v2 == v1 && i2 < i1)) { s_val[j] = v2; s_idx[j] = i2; }
    }
    __syncthreads();
  }
  if (j == 0) s_pkidx = s_idx[0];
  __syncthreads();

  const int knew = mask[n] ? qk : s_pkidx;
  const float dkl = -logf(s_pk[qk]);

  // ---- z_new = C[knew], kl = 1.25 * || gt - z_new || ----
  float diff2 = 0.0f;
  if (j < DZ) {
    const float cv = C[(size_t)knew * DZ + j];
    out_z[(size_t)n * DZ + j] = cv;
    const float df = s_gt[j] - cv;
    diff2 = df * df;
  }
  s_val[j] = diff2;
  __syncthreads();
  for (int off = KK / 2; off > 0; off >>= 1) {
    if (j < off) s_val[j] += s_val[j + off];
    __syncthreads();
  }
  if (j == 0) {
    const float dist = sqrtf(s_val[0]);
    out_kl[n]  = (1.0f + BETA_F) * dist + dkl;
    out_dkl[n] = dkl;
  }
}

// ---------------------------------------------------------------------------
extern "C" void kernel_launch(void* const* d_in, const int* in_sizes, int n_in,
                              void* d_out, int out_size, void* d_ws, size_t ws_size,
                              hipStream_t stream) {
  const float* z_sample   = (const float*)d_in[1];
  const float* k_sample   = (const float*)d_in[2];
  const float* xt         = (const float*)d_in[3];
  const unsigned char* mk = (const unsigned char*)d_in[4];   // bool mask
  const float* transition = (const float*)d_in[5];
  const float* start_pk   = (const float*)d_in[6];
  const float* W1 = (const float*)d_in[7];
  const float* b1 = (const float*)d_in[8];
  const float* W2 = (const float*)d_in[9];
  const float* b2 = (const float*)d_in[10];
  const float* W3 = (const float*)d_in[11];
  const float* b3 = (const float*)d_in[12];
  const float* C  = (const float*)d_in[13];
  const int* seedp = (const int*)d_in[14];

  float* ws = (float*)d_ws;
  float* X1 = ws;                                  // [N,192]
  float* H1 = X1 + (size_t)N_TOK * KIN;            // [N,256]
  float* H2 = H1 + (size_t)N_TOK * HH;             // [N,256]
  float* GT = H2 + (size_t)N_TOK * HH;             // [N,64]

  float* out     = (float*)d_out;
  float* out_z   = out;                            // [N,64]
  float* out_kl  = out_z + (size_t)N_TOK * DZ;     // [N]
  float* out_dkl = out_kl + N_TOK;                 // [N]
  float* out_qk  = out_dkl + N_TOK;                // [N,K]

  prep_x1<<<(N_TOK * KIN + 255) / 256, 256, 0, stream>>>(z_sample, xt, X1);

  // one wave per 16x64 strip; 4 waves (128 threads) per block
  gemm_wmma<KIN, HH, true ><<<(N_TOK / 16) * (HH / 64) / 4, 128, 0, stream>>>(X1, W1, b1, H1);
  gemm_wmma<HH,  HH, true ><<<(N_TOK / 16) * (HH / 64) / 4, 128, 0, stream>>>(H1, W2, b2, H2);
  gemm_wmma<HH,  DZ, false><<<(N_TOK / 16) * (DZ / 64) / 4, 128, 0, stream>>>(H2, W3, b3, GT);

  fused_tail<<<N_TOK, KK, 0, stream>>>(GT, k_sample, transition, start_pk, mk, C,
                                       seedp, out_z, out_kl, out_dkl, out_qk);
}